// real_space_DNN_71674414235683
// MI455X (gfx1250) — compile-verified
//
#include <hip/hip_runtime.h>
#include <cmath>

typedef __attribute__((ext_vector_type(2))) float v2f;
typedef __attribute__((ext_vector_type(8))) float v8f;

#define NPIX 96
#define NN   (NPIX * NPIX)      // 9216 pixels per plane
#define NB   16                 // batch == WMMA M dimension
#define NNEU 24                 // phase grid (nearest-upsampled x4)
#define KSPLIT 16               // waves per tile, splitting the K reduction
#define KSLICE (NN / KSPLIT)    // 576 K-values per wave
#define TWO_PI_F 6.28318530717958647692f

// D = A(16x4 f32) * B(4x16 f32) + C(16x16 f32), one matrix per wave32.
__device__ __forceinline__ v8f wmma4_f32(v2f a, v2f b, v8f c) {
    // 8 args: (neg_a, A, neg_b, B, c_mod, C, reuse_a, reuse_b)
    return __builtin_amdgcn_wmma_f32_16x16x4_f32(
        false, a, false, b, (short)0, c, false, false);
}

// Non-temporal 8-byte load: K streams once per hop and overflows L2.
__device__ __forceinline__ float2 ldnt_f2(const float2* p) {
    unsigned long long v =
        __builtin_nontemporal_load((const unsigned long long*)p);
    union { unsigned long long u; float2 f; } c;
    c.u = v;
    return c.f;
}

// One propagation hop: out[16, NN] = A[16, NN] @ Bm[NN, NN]  (complex GEMM),
// then either phase-rotate (intermediate layer) or square-magnitude (final).
// One block per 16-column output tile; 16 waves split the K=9216 reduction
// and combine through LDS.
template <bool A_REAL, bool FINAL>
__global__ __launch_bounds__(512) void hop_kernel(
    const float* __restrict__ A,    // A_REAL: [16,NN] f32, else [16,NN] cplx interleaved
    const float* __restrict__ Bm,   // [NN,NN] complex64 interleaved (re,im)
    const float* __restrict__ ph,   // [24,24] raw phase params (unused if FINAL)
    float* __restrict__ out)        // FINAL: [16,NN] f32, else [16,NN] cplx interleaved
{
    __shared__ float red[KSPLIT * 512];   // 32 KB: per-wave (re,im) partial tiles

    const int tid  = threadIdx.x;
    const int wv   = tid >> 5;            // K-slice this wave owns (0..15)
    const int lane = tid & 31;
    const int half = lane >> 4;           // 0: K=k0,k0+1 / M=r ; 1: K=k0+2,k0+3 / M=r+8
    const int lcol = lane & 15;
    const int n0   = blockIdx.x * 16;     // output tile base column
    const int n    = n0 + lcol;           // B/C/D column this lane owns

    v8f acc_rr = {};  // Re(A)*Re(B)
    v8f acc_ii = {};  // Im(A)*Im(B)
    v8f acc_ri = {};  // Re(A)*Im(B)
    v8f acc_ir = {};  // Im(A)*Re(B)

    const float2* __restrict__ Bc = (const float2*)Bm;
    const float2* __restrict__ Ac = (const float2*)A;
    const long long arow = (long long)lcol * NN;   // A row for this lane (M = lcol)

    const int kbeg = wv * KSLICE;
    const int kend = kbeg + KSLICE;       // 144 k-steps of 4 per wave

    #pragma unroll 8
    for (int k0 = kbeg; k0 < kend; k0 += 4) {
        const int kb = k0 + 2 * half;     // this lane's K pair

        // B rows kb, kb+1 at column n: each instr = 128B/half-wave, NT (no reuse)
        float2 b0 = ldnt_f2(&Bc[(long long)kb * NN + n]);
        float2 b1 = ldnt_f2(&Bc[(long long)(kb + 1) * NN + n]);
        v2f br = {b0.x, b1.x};
        v2f bi = {b0.y, b1.y};

        v2f ar, ai;
        if (A_REAL) {
            float2 a01 = *(const float2*)(A + arow + kb);  // 2 consecutive reals
            ar = (v2f){a01.x, a01.y};
            ai = (v2f){0.0f, 0.0f};
        } else {
            float2 a0 = Ac[arow + kb];
            float2 a1 = Ac[arow + kb + 1];
            ar = (v2f){a0.x, a1.x};
            ai = (v2f){a0.y, a1.y};
        }

        acc_rr = wmma4_f32(ar, br, acc_rr);
        acc_ri = wmma4_f32(ar, bi, acc_ri);
        if (!A_REAL) {
            acc_ii = wmma4_f32(ai, bi, acc_ii);
            acc_ir = wmma4_f32(ai, br, acc_ir);
        }
    }

    // Pre-combine to (re, im) and publish this wave's partial tile.
    #pragma unroll
    for (int r = 0; r < 8; ++r) {
        red[wv * 512 + lane * 16 + 2 * r]     = acc_rr[r] - acc_ii[r];
        red[wv * 512 + lane * 16 + 2 * r + 1] = acc_ri[r] + acc_ir[r];
    }
    __syncthreads();

    // 256 threads: sum the 16 K-slice partials and run the fused epilogue.
    if (tid < 256) {
        const int l = tid >> 3;           // logical lane 0..31 of the C layout
        const int r = tid & 7;            // accumulator VGPR slot 0..7
        float re = 0.0f, im = 0.0f;
        #pragma unroll
        for (int w = 0; w < KSPLIT; ++w) {
            re += red[w * 512 + l * 16 + 2 * r];
            im += red[w * 512 + l * 16 + 2 * r + 1];
        }

        const int m  = r + 8 * (l >> 4);  // batch row (C layout: M = r + 8*half)
        const int nn = n0 + (l & 15);     // output pixel

        if (FINAL) {
            out[(long long)m * NN + nn] = re * re + im * im;   // intensity
        } else {
            const int i = nn / NPIX;
            const int j = nn - i * NPIX;
            const float p = ph[(i >> 2) * NNEU + (j >> 2)];    // nearest x4 upsample
            const float s = 1.0f / (1.0f + __expf(-p));        // sigmoid
            float sn, cs;
            __sincosf(TWO_PI_F * s, &sn, &cs);
            float2 o;
            o.x = re * cs - im * sn;                           // field * exp(i*ang)
            o.y = re * sn + im * cs;
            ((float2*)out)[(long long)m * NN + nn] = o;
        }
    }
}

extern "C" void kernel_launch(void* const* d_in, const int* in_sizes, int n_in,
                              void* d_out, int out_size, void* d_ws, size_t ws_size,
                              hipStream_t stream) {
    (void)in_sizes; (void)n_in; (void)out_size; (void)ws_size;

    const float* x     = (const float*)d_in[0];   // [16,96,96] f32
    const float* phase = (const float*)d_in[1];   // [5,24,24] f32
    const float* K0    = (const float*)d_in[2];   // [9216,9216] complex64
    const float* K     = (const float*)d_in[3];   // [9216,9216] complex64
    float*       out   = (float*)d_out;           // [16,96,96] f32

    // Ping-pong complex field buffers in workspace: 2 x 16*9216 complex64 = 2.36 MB.
    float* buf0 = (float*)d_ws;
    float* buf1 = buf0 + (size_t)NB * NN * 2;

    const dim3 grid(NN / 16);   // 576 blocks, one output tile each
    const dim3 block(512);      // 16 wave32s splitting the K reduction

    // layer 0: real input through K0, phase[0]
    hop_kernel<true, false><<<grid, block, 0, stream>>>(x, K0, phase + 0 * NNEU * NNEU, buf0);
    // layers 1..4: complex field through K, phase[n]
    hop_kernel<false, false><<<grid, block, 0, stream>>>(buf0, K, phase + 1 * NNEU * NNEU, buf1);
    hop_kernel<false, false><<<grid, block, 0, stream>>>(buf1, K, phase + 2 * NNEU * NNEU, buf0);
    hop_kernel<false, false><<<grid, block, 0, stream>>>(buf0, K, phase + 3 * NNEU * NNEU, buf1);
    hop_kernel<false, false><<<grid, block, 0, stream>>>(buf1, K, phase + 4 * NNEU * NNEU, buf0);
    // final propagation to detector + intensity
    hop_kernel<false, true><<<grid, block, 0, stream>>>(buf0, K, nullptr, out);
}